// Conv2DUF_71021579207337
// MI455X (gfx1250) — compile-verified
//
#include <hip/hip_runtime.h>

typedef __attribute__((ext_vector_type(2))) float v2f;
typedef __attribute__((ext_vector_type(4))) float v4f;
typedef __attribute__((ext_vector_type(8))) float v8f;
typedef __attribute__((ext_vector_type(4))) unsigned int v4u;
typedef __attribute__((ext_vector_type(4))) int v4i;
typedef __attribute__((ext_vector_type(8))) int v8i;

#define C_IN   128
#define C_OUT  256
#define IMG_H  56
#define IMG_W  56
#define HW     (IMG_H * IMG_W)   // 3136
#define BATCH  32
#define CHUNK  32                // input channels per K-chunk
#define NCHUNK 36                // 9 taps x (128/32)

// Block tile: 64 pixels (M) x 128 out-channels (N), 256 threads = 8 waves.
// Wave grid 2(M) x 4(N); each wave computes a 32x32 tile = 4 f32 accumulators.
// Pipeline: double-buffered LDS; chunk t+1 is staged (TDM weight tile + global
// im2col loads) while chunk t runs 32 x V_WMMA_F32_16X16X4_F32.
__global__ __launch_bounds__(256)
void conv2duf_wmma_f32(const float* __restrict__ x,      // [B, C_IN, H, W]
                       const float* __restrict__ wmat,   // [C_IN*9, C_OUT] row = c*9 + kh*3 + kw
                       const float* __restrict__ bias,   // [C_OUT]
                       float* __restrict__ out) {        // [B, C_OUT, H, W]
    __shared__ float a_lds[2][64 * CHUNK];    // [pix][ch]  2 x 8 KB
    __shared__ float b_lds[2][CHUNK * 128];   // [k][n]     2 x 16 KB (TDM layout)

    const int tid   = threadIdx.x;
    const int lane  = tid & 31;
    const int wid   = tid >> 5;
    const int mBase2 = (wid & 1) * 32;    // wave M sub-tile within 64
    const int nSub   = (wid >> 1) * 32;   // wave N sub-tile within 128

    const int mBlock = blockIdx.x;            // 0..1567
    const int nBase  = blockIdx.y * 128;      // 0 or 128
    const int b      = mBlock / 49;           // 3136/64 = 49 tiles per image
    const int pBase  = (mBlock % 49) * 64;    // tile never crosses an image

    // ---- A staging map: thread -> (pixel, 8 consecutive channels)
    const int aPix = tid >> 2;                // 0..63
    const int aCh8 = (tid & 3) * 8;           // 0,8,16,24
    const int p    = pBase + aPix;
    const int ph   = p / IMG_W;
    const int pw   = p % IMG_W;
    const float* xb = x + (size_t)b * C_IN * HW;

    // ---- TDM constants (wave-uniform)
    const int widS = __builtin_amdgcn_readfirstlane(wid);
    const unsigned long long wbase = (unsigned long long)(const void*)wmat;

    // ---- per-lane fragment coordinates
    const int lane15 = lane & 15;
    const int kHalf  = (lane >> 4) * 2;   // lanes 0-15 -> K+0/1, lanes 16-31 -> K+2/3

    v8f acc00 = {0,0,0,0,0,0,0,0};
    v8f acc01 = {0,0,0,0,0,0,0,0};
    v8f acc10 = {0,0,0,0,0,0,0,0};
    v8f acc11 = {0,0,0,0,0,0,0,0};

    // ---- stage chunk t into buffer buf: TDM for B (4 k-rows per wave),
    //      predicated strided global loads + 2 x ds_store_b128 for A.
    auto stage = [&](int t, int buf) {
        const int pos = t >> 2;               // tap 0..8 (uniform)
        const int c0  = (t & 3) * CHUNK;      // channel base (uniform)
        // B tile via Tensor Data Mover
        {
            const unsigned int ldsAddr =
                (unsigned int)(size_t)(&b_lds[buf][0]) +
                (unsigned int)(widS * 4 * 128 * 4);
            const unsigned long long ga =
                wbase + (unsigned long long)(((c0 + widS * 4) * 9 + pos) * C_OUT + nBase) * 4ull;
            v4u g0;
            g0.x = 1u;                                   // count=1, user descriptor
            g0.y = ldsAddr;                              // LDS byte address
            g0.z = (unsigned int)(ga & 0xffffffffull);   // global_addr[31:0]
            g0.w = (unsigned int)((ga >> 32) & 0x1ffffffull) | (2u << 30); // [56:32] | type=2
            v8i g1;
            g1.s0 = 0x00020000;            // data_size=2 (4 bytes), mask/pad/iterate=0
            g1.s1 = 0x00000000;            // tensor_dim0[15:0]=0
            g1.s2 = 0x00000010;            // tensor_dim0=1M, tensor_dim1 lo=0
            g1.s3 = 0x00800010;            // tensor_dim1=1M, tile_dim0=128 floats
            g1.s4 = 0x00000004;            // tile_dim1=4 rows, tile_dim2=0
            g1.s5 = 9 * C_OUT;             // tensor_dim0_stride = 2304 floats
            g1.s6 = 0;
            g1.s7 = 0;
            v4i gz4 = {0, 0, 0, 0};
            v8i gz8 = {0, 0, 0, 0, 0, 0, 0, 0};
            __builtin_amdgcn_tensor_load_to_lds(g0, g1, gz4, gz4, gz8, 0);
        }
        // A tile (zero-padded im2col)
        {
            const int kh = pos / 3, kw = pos % 3;
            const int ih = ph + kh - 1;
            const int iw = pw + kw - 1;
            const bool valid = (ih >= 0) & (ih < IMG_H) & (iw >= 0) & (iw < IMG_W);
            const int spat = ih * IMG_W + iw;
            float v[8] = {0.f, 0.f, 0.f, 0.f, 0.f, 0.f, 0.f, 0.f};
            if (valid) {
                const float* src = xb + (size_t)(c0 + aCh8) * HW + spat;
                #pragma unroll
                for (int i = 0; i < 8; ++i) v[i] = src[(size_t)i * HW];
            }
            v4f lo = {v[0], v[1], v[2], v[3]};
            v4f hi = {v[4], v[5], v[6], v[7]};
            *(v4f*)&a_lds[buf][aPix * CHUNK + aCh8]     = lo;
            *(v4f*)&a_lds[buf][aPix * CHUNK + aCh8 + 4] = hi;
        }
    };

    stage(0, 0);   // prologue

    for (int t = 0; t < NCHUNK; ++t) {
        const int buf = t & 1;
        __builtin_amdgcn_s_wait_tensorcnt(0);   // this wave's TDM rows for buf landed
        __syncthreads();                        // all staging of buf done; buf^1 free
        if (t + 1 < NCHUNK) stage(t + 1, buf ^ 1);

        const float* aB = &a_lds[buf][0];
        const float* bB = &b_lds[buf][0];
        #pragma unroll
        for (int kk = 0; kk < CHUNK / 4; ++kk) {
            const int kb = kk * 4 + kHalf;
            v2f a0 = *(const v2f*)&aB[(mBase2 + lane15) * CHUNK + kb];
            v2f a1 = *(const v2f*)&aB[(mBase2 + 16 + lane15) * CHUNK + kb];
            v2f b0, b1;
            b0.x = bB[kb * 128 + nSub + lane15];
            b0.y = bB[(kb + 1) * 128 + nSub + lane15];
            b1.x = bB[kb * 128 + nSub + 16 + lane15];
            b1.y = bB[(kb + 1) * 128 + nSub + 16 + lane15];
            acc00 = __builtin_amdgcn_wmma_f32_16x16x4_f32(false, a0, false, b0,
                                                          (short)0, acc00, false, false);
            acc01 = __builtin_amdgcn_wmma_f32_16x16x4_f32(false, a0, false, b1,
                                                          (short)0, acc01, false, false);
            acc10 = __builtin_amdgcn_wmma_f32_16x16x4_f32(false, a1, false, b0,
                                                          (short)0, acc10, false, false);
            acc11 = __builtin_amdgcn_wmma_f32_16x16x4_f32(false, a1, false, b1,
                                                          (short)0, acc11, false, false);
        }
    }

    // ---- epilogue: bias + store; lane owns one column per j, 8 consecutive pixels per acc
    const int mHalf = (lane >> 4) * 8;
    #pragma unroll
    for (int j = 0; j < 2; ++j) {
        const int nGlob = nBase + nSub + j * 16 + lane15;
        const float bv  = bias[nGlob];
        v8f aL = (j == 0) ? acc00 : acc01;   // rows mBase2 + 0..15
        v8f aH = (j == 0) ? acc10 : acc11;   // rows mBase2 + 16..31
        float* op = out + (size_t)b * C_OUT * HW + (size_t)nGlob * HW
                        + pBase + mBase2 + mHalf;
        v4f s0 = {aL[0] + bv, aL[1] + bv, aL[2] + bv, aL[3] + bv};
        v4f s1 = {aL[4] + bv, aL[5] + bv, aL[6] + bv, aL[7] + bv};
        v4f s2 = {aH[0] + bv, aH[1] + bv, aH[2] + bv, aH[3] + bv};
        v4f s3 = {aH[4] + bv, aH[5] + bv, aH[6] + bv, aH[7] + bv};
        *(v4f*)(op)      = s0;   // global_store_b128
        *(v4f*)(op + 4)  = s1;
        *(v4f*)(op + 16) = s2;
        *(v4f*)(op + 20) = s3;
    }
}

extern "C" void kernel_launch(void* const* d_in, const int* in_sizes, int n_in,
                              void* d_out, int out_size, void* d_ws, size_t ws_size,
                              hipStream_t stream) {
    const float* x    = (const float*)d_in[0];   // [32,128,56,56]
    const float* wmat = (const float*)d_in[1];   // [1152,256]
    const float* bias = (const float*)d_in[2];   // [256]
    float* out        = (float*)d_out;           // [32,256,56,56]

    dim3 grid(BATCH * HW / 64, C_OUT / 128);     // (1568, 2)
    dim3 block(256);
    conv2duf_wmma_f32<<<grid, block, 0, stream>>>(x, wmat, bias, out);
}